// PropagationBlock_4243427689050
// MI455X (gfx1250) — compile-verified
//
#include <hip/hip_runtime.h>

typedef float v2f __attribute__((ext_vector_type(2)));
typedef float v8f __attribute__((ext_vector_type(8)));
typedef unsigned int u32x4 __attribute__((ext_vector_type(4)));
typedef int i32x8 __attribute__((ext_vector_type(8)));
typedef int i32x4 __attribute__((ext_vector_type(4)));

#define NC 512
#define M 1024
#define KS 24
#define NA 60
#define CO 128
#define NKA (KS*NA)          // 1440
#define KATILES (NKA/16)     // 90
#define J (NC*NA)            // 30720
#define FEATN (2*CO*J)       // 7864320

// ---------------------------------------------------------------------------
// TDM: 1-D tensor DMA of nelem f32 from global to LDS (D# per CDNA5 ISA ch.8).
// group0: count=1 | lds_addr | global_addr[56:0] | type=2
// group1: data_size=4B (bits 17:16); tensor_dim0 at bits 79:48; tile_dim0 at
//         bits 127:112; tile_dim1/2 = 0 (unused -> 1-D); strides unused.
// groups 2/3: zero; 6-arg toolchain form with extra i32x8; cpol=0.
// ---------------------------------------------------------------------------
__device__ __forceinline__ void tdm_load_1d_f32(const void* gaddr,
                                                unsigned lds_off,
                                                unsigned nelem)
{
    unsigned long long ga = (unsigned long long)gaddr;
    u32x4 g0;
    g0[0] = 1u;                                           // count=1, user mode
    g0[1] = lds_off;                                      // lds_addr (bytes)
    g0[2] = (unsigned)(ga & 0xFFFFFFFFu);                 // global_addr[31:0]
    g0[3] = (unsigned)((ga >> 32) & 0x01FFFFFFu) | (2u << 30); // addr[56:32], type=2
    i32x8 g1;
    g1[0] = (int)(2u << 16);                              // data_size = 2 (4 bytes)
    g1[1] = (int)((nelem & 0xFFFFu) << 16);               // tensor_dim0[15:0] @ bits 63:48
    g1[2] = (int)((nelem >> 16) & 0xFFFFu);               // tensor_dim0[31:16] @ bits 79:64
    g1[3] = (int)((nelem & 0xFFFFu) << 16);               // tile_dim0 @ bits 127:112
    g1[4] = 0; g1[5] = 0; g1[6] = 0; g1[7] = 0;
    i32x4 z4 = {0, 0, 0, 0};
    i32x8 z8 = {0, 0, 0, 0, 0, 0, 0, 0};
    __builtin_amdgcn_tensor_load_to_lds(g0, g1, z4, z4, z8, 0);
}

// ---------------------------------------------------------------------------
// Kernel 1: wts[b][k][n*60+a] = (1/(nnctn+1)) * sum_m mask * exp(-|diff-kp|^2/sigma)
// One wave = one (b,n) cloud point x one 16-wide (k,a) tile; 64 m-tiles.
//   A = (25*dx, 25*dy, 25*dz, -12.5*d2)  (mask folded: 4th comp = -1e30)
//   B = (kx, ky, kz, 1)      => dot = (2*diff.kp - d2)/sigma
//   w = exp(-|kp|^2/sigma) * exp(dot)
// frag (12 KB) staged into LDS by TDM once per block.
// ---------------------------------------------------------------------------
__global__ void k1_wts(const float* __restrict__ frag,
                       const float* __restrict__ clouds,
                       const float* __restrict__ kernels,
                       float* __restrict__ wts)
{
    __shared__ float shfrag[M * 3];                       // 12288 B

    if ((threadIdx.x >> 5) == 0) {                        // wave 0 issues the DMA
        tdm_load_1d_f32(frag, (unsigned)(unsigned long long)(void*)shfrag, M * 3);
        __builtin_amdgcn_s_wait_tensorcnt(0);
    }
    __syncthreads();

    const int wave = blockIdx.x * (blockDim.x >> 5) + (threadIdx.x >> 5);
    const int lane = threadIdx.x & 31;
    const int katile = wave % KATILES;
    const int bn = wave / KATILES;
    const int b = bn >> 9;
    const int n = bn & (NC - 1);

    const float cx = clouds[b * 3 * NC + 0 * NC + n];
    const float cy = clouds[b * 3 * NC + 1 * NC + n];
    const float cz = clouds[b * 3 * NC + 2 * NC + n];

    const int f = katile * 16 + (lane & 15);              // flat (k,a) index
    const float kx = kernels[f * 3 + 0];
    const float ky = kernels[f * 3 + 1];
    const float kz = kernels[f * 3 + 2];
    v2f bmat;
    bmat.x = (lane < 16) ? kx : kz;                       // K0 / K2
    bmat.y = (lane < 16) ? ky : 1.0f;                     // K1 / K3
    const float kexe = __expf(-12.5f * (kx * kx + ky * ky + kz * kz));

    float s = 0.0f;
    int nn = 0;
    for (int mt = 0; mt < M / 16; ++mt) {
        const int ml = mt * 16 + (lane & 15);
        const float fx = shfrag[ml * 3 + 0];
        const float fy = shfrag[ml * 3 + 1];
        const float fz = shfrag[ml * 3 + 2];
        const float dx = fx - cx, dy = fy - cy, dz = fz - cz;
        const float d2 = dx * dx + dy * dy + dz * dz;
        const bool within = d2 < 0.16f;                   // RADIUS^2
        nn += within ? 1 : 0;
        if (__any((int)within)) {                         // uniform: EXEC all-ones inside
            v2f a;
            a.x = (lane < 16) ? 25.0f * dx : 25.0f * dz;                        // K0 / K2
            a.y = (lane < 16) ? 25.0f * dy : (within ? -12.5f * d2 : -1.0e30f); // K1 / K3
            v8f c = {};
            c = __builtin_amdgcn_wmma_f32_16x16x4_f32(false, a, false, bmat,
                                                      (short)0, c, false, false);
            s += __expf(c[0]) + __expf(c[1]) + __expf(c[2]) + __expf(c[3]) +
                 __expf(c[4]) + __expf(c[5]) + __expf(c[6]) + __expf(c[7]);
        }
    }

    // nn duplicated in both half-waves; sum over 32 lanes = 2 * count
    for (int off = 16; off > 0; off >>= 1) nn += __shfl_xor(nn, off, 32);
    const float invnn = 1.0f / (0.5f * (float)nn + 1.0f);

    // combine row-halves of the C layout: lane L and L+16 hold the same column
    s += __shfl_xor(s, 16, 32);

    if (lane < 16) {
        const int k = f / NA;
        const int aI = f - k * NA;
        wts[(b * KS + k) * J + n * NA + aI] = kexe * s * invnn;
    }
}

// ---------------------------------------------------------------------------
// Kernel 2: feats[b][c][j] = sum_k W[c][k] * wts[b][k][j]   (K=24 -> 6 WMMA steps)
// W (12 KB) staged into LDS by TDM once per block.
// ---------------------------------------------------------------------------
__global__ void k2_gemm(const float* __restrict__ Wm,
                        const float* __restrict__ wts,
                        float* __restrict__ feats)
{
    __shared__ float shW[CO * KS];                        // 12288 B

    if ((threadIdx.x >> 5) == 0) {
        tdm_load_1d_f32(Wm, (unsigned)(unsigned long long)(void*)shW, CO * KS);
        __builtin_amdgcn_s_wait_tensorcnt(0);
    }
    __syncthreads();

    const int wave = blockIdx.x * (blockDim.x >> 5) + (threadIdx.x >> 5);
    const int lane = threadIdx.x & 31;
    const int jt = wave % (J / 16);
    const int rest = wave / (J / 16);
    const int ct = rest & 7;
    const int b = rest >> 3;

    const int row = ct * 16 + (lane & 15);
    const int j = jt * 16 + (lane & 15);
    const int khalf = (lane < 16) ? 0 : 2;
    const float* wb = wts + b * KS * J;

    __builtin_prefetch(wb + j, 0, 3);                     // global_prefetch_b8

    v8f c = {};
    #pragma unroll
    for (int kk = 0; kk < 6; ++kk) {
        const int k0 = kk * 4 + khalf;
        v2f a, bm;
        a.x = shW[row * KS + k0];
        a.y = shW[row * KS + k0 + 1];
        bm.x = wb[k0 * J + j];
        bm.y = wb[(k0 + 1) * J + j];
        c = __builtin_amdgcn_wmma_f32_16x16x4_f32(false, a, false, bm,
                                                  (short)0, c, false, false);
    }

    float* out = feats + b * CO * J;
    const int rbase = ct * 16 + ((lane < 16) ? 0 : 8);
    #pragma unroll
    for (int v = 0; v < 8; ++v)
        out[(rbase + v) * J + j] = c[v];
}

// ---------------------------------------------------------------------------
// Kernel 3: per (b,c): mean and rstd over 30720 elements
// ---------------------------------------------------------------------------
__global__ void k3_stats(const float* __restrict__ feats, float* __restrict__ stats)
{
    __shared__ float sh[256];
    __shared__ float sh2[256];
    const int bc = blockIdx.x;                 // 0..255
    const float* p = feats + (size_t)bc * J;
    float sum = 0.f, sq = 0.f;
    for (int i = threadIdx.x; i < J; i += 256) {
        float v = p[i];
        sum += v; sq += v * v;
    }
    sh[threadIdx.x] = sum; sh2[threadIdx.x] = sq;
    __syncthreads();
    for (int off = 128; off > 0; off >>= 1) {
        if ((int)threadIdx.x < off) {
            sh[threadIdx.x] += sh[threadIdx.x + off];
            sh2[threadIdx.x] += sh2[threadIdx.x + off];
        }
        __syncthreads();
    }
    if (threadIdx.x == 0) {
        const float inv = 1.0f / (float)J;
        float mean = sh[0] * inv;
        float var = sh2[0] * inv - mean * mean;
        stats[bc * 2 + 0] = mean;
        stats[bc * 2 + 1] = rsqrtf(var + 1e-5f);
    }
}

// ---------------------------------------------------------------------------
// Kernel 4: in-place layernorm + relu on the feat region of d_out
// ---------------------------------------------------------------------------
__global__ void k4_norm(float* __restrict__ feat, const float* __restrict__ stats)
{
    const int idx = blockIdx.x * 256 + threadIdx.x;   // < FEATN
    const int bc = idx / J;
    const float mean = stats[bc * 2 + 0];
    const float rstd = stats[bc * 2 + 1];
    float v = (feat[idx] - mean) * rstd;
    feat[idx] = v > 0.f ? v : 0.f;
}

// ---------------------------------------------------------------------------
// Kernel 5: copy passthrough outputs (clouds, anchors) into d_out
// ---------------------------------------------------------------------------
__global__ void k5_copy(const float* __restrict__ clouds,
                        const float* __restrict__ anchors,
                        float* __restrict__ out)
{
    const int i = blockIdx.x * 256 + threadIdx.x;
    if (i < 2 * 3 * NC) out[i] = clouds[i];
    if (i < NA * 9) out[2 * 3 * NC + FEATN + i] = anchors[i];
}

extern "C" void kernel_launch(void* const* d_in, const int* in_sizes, int n_in,
                              void* d_out, int out_size, void* d_ws, size_t ws_size,
                              hipStream_t stream)
{
    const float* frag    = (const float*)d_in[0];
    const float* clouds  = (const float*)d_in[1];
    const float* kernels = (const float*)d_in[2];
    const float* Wm      = (const float*)d_in[3];
    const float* anchors = (const float*)d_in[4];
    float* out  = (float*)d_out;
    float* wts  = (float*)d_ws;                       // 2*24*30720 f32 = 5.9 MB
    float* stats = wts + 2 * KS * J;                  // 512 f32
    float* feat = out + 2 * 3 * NC;                   // feat region of d_out

    // kernel1: 2*512 cloud pts * 90 ka-tiles = 92160 waves, 4 waves/block
    k1_wts<<<(2 * NC * KATILES) / 4, 128, 0, stream>>>(frag, clouds, kernels, wts);
    // kernel2: 2 * 8 ctiles * 1920 jtiles = 30720 waves, 4 waves/block
    k2_gemm<<<(2 * 8 * (J / 16)) / 4, 128, 0, stream>>>(Wm, wts, feat);
    k3_stats<<<2 * CO, 256, 0, stream>>>(feat, stats);
    k4_norm<<<FEATN / 256, 256, 0, stream>>>(feat, stats);
    k5_copy<<<(2 * 3 * NC + 255) / 256, 256, 0, stream>>>(clouds, anchors, out);
}